// MosaicNGramCacheLogitsLayer_4080218931222
// MI455X (gfx1250) — compile-verified
//
#include <hip/hip_runtime.h>
#include <stdint.h>

// MosaicNGramCacheLogitsLayer for MI455X (gfx1250).
// out = logits + sparse n-gram-cache deltas.
// Memory-bound: ~1.65 GB traffic -> ~71 us at 23.3 TB/s. Copy uses the CDNA5
// async global<->LDS path (ASYNCcnt). The "sequential" scan is decomposed into
// independent per-hash-slot chains processed in parallel (state in registers).

#define NGRAM 4
#define TOP_M 4
#define TABLE 65536
#define BASEH 1315423911ull
#define BT 256
#define KU 8  // async B128 ops in flight per thread per wait

typedef unsigned long long u64;

#if defined(__HIP_DEVICE_COMPILE__) && defined(__gfx1250__)
#  if __has_builtin(__builtin_amdgcn_global_load_async_to_lds_b128) && \
      __has_builtin(__builtin_amdgcn_global_store_async_from_lds_b128)
#    define USE_ASYNC_COPY 1
#  endif
#  if __has_builtin(__builtin_amdgcn_s_wait_asynccnt)
#    define WAIT_ASYNC() __builtin_amdgcn_s_wait_asynccnt(0)
#  else
#    define WAIT_ASYNC() asm volatile("s_wait_asynccnt 0x0" ::: "memory")
#  endif
#endif

// ---------- K3 (defined first so the disasm snippet shows it):
// bulk copy logits -> out via async global->LDS->global streaming ----------
__global__ void k_copy(const float* __restrict__ inF, float* __restrict__ outF,
                       long long N) {
  long long n4 = N >> 2;
  const float4* __restrict__ in4 = (const float4*)inF;
  float4* __restrict__ out4 = (float4*)outF;
  int tid = threadIdx.x;
#if defined(USE_ASYNC_COPY)
  __shared__ float4 stage[BT * KU];  // 32 KB; each thread owns KU private slots
  typedef int v4i __attribute__((ext_vector_type(4)));
  typedef __attribute__((address_space(1))) v4i dev_v4i;  // clang: "__device__"
  typedef __attribute__((address_space(3))) v4i lds_v4i;  // clang: "__shared__"
  long long chunk = (long long)blockIdx.x * (BT * KU);
  long long cstride = (long long)gridDim.x * (BT * KU);
  for (; chunk < n4; chunk += cstride) {
#pragma unroll
    for (int k = 0; k < KU; ++k) {
      long long i = chunk + (long long)k * BT + tid;
      if (i < n4)
        __builtin_amdgcn_global_load_async_to_lds_b128(
            (dev_v4i*)(in4 + i), (lds_v4i*)&stage[k * BT + tid], 0, 0);
    }
    WAIT_ASYNC();  // per-wave: our lanes' loads have landed in LDS
#pragma unroll
    for (int k = 0; k < KU; ++k) {
      long long i = chunk + (long long)k * BT + tid;
      if (i < n4)
        __builtin_amdgcn_global_store_async_from_lds_b128(
            (dev_v4i*)(out4 + i), (lds_v4i*)&stage[k * BT + tid], 0, 0);
    }
    WAIT_ASYNC();  // stores drained before LDS slots are reused
  }
#else
  long long gid = (long long)blockIdx.x * BT + tid;
  long long gstride = (long long)gridDim.x * BT;
  for (long long i = gid; i < n4; i += gstride) out4[i] = in4[i];
#endif
  // scalar tail (N % 4), empty for this problem size
  long long tgid = (long long)blockIdx.x * BT + tid;
  long long tstride = (long long)gridDim.x * BT;
  for (long long i = (n4 << 2) + tgid; i < N; i += tstride) outF[i] = inF[i];
}

// ---------- K0: init first-seen table ----------
__global__ void k_init_first(int* __restrict__ first) {
  int s = blockIdx.x * blockDim.x + threadIdx.x;
  if (s < TABLE) first[s] = 0x7FFFFFFF;
}

// ---------- K1: per-t fingerprint/slot/key; zero emission arrays ----------
__global__ void k_prep(const int* __restrict__ ids, int* __restrict__ slot,
                       u64* __restrict__ key, int* __restrict__ ntp1,
                       int* __restrict__ idxs, float* __restrict__ vals,
                       int* __restrict__ first, int T) {
  int t = blockIdx.x * blockDim.x + threadIdx.x;
  if (t >= T) return;
#pragma unroll
  for (int m = 0; m < TOP_M; ++m) {
    idxs[t * TOP_M + m] = 0;
    vals[t * TOP_M + m] = 0.0f;
  }
  ntp1[t] = (t < T - 1) ? (ids[t + 1] + 1) : 0;
  if (t >= NGRAM - 1) {
    // fp_t = sum_{j=0..3} (id_{t-j}+1) * B^j  (mod 2^64), Horner form
    u64 fp = (u64)(unsigned)(ids[t - 3] + 1);
    fp = fp * BASEH + (u64)(unsigned)(ids[t - 2] + 1);
    fp = fp * BASEH + (u64)(unsigned)(ids[t - 1] + 1);
    fp = fp * BASEH + (u64)(unsigned)(ids[t] + 1);
    int s = (int)(fp & (u64)(TABLE - 1));
    slot[t] = s;
    key[t] = fp + 1ull;  // want_key
    atomicMin(&first[s], t);
  } else {
    slot[t] = -1;
    key[t] = 0ull;
  }
}

// ---------- K2: one leader thread per touched slot replays its chain ----------
__global__ void k_chains(const int* __restrict__ slot, const u64* __restrict__ key,
                         const int* __restrict__ ntp1, int* __restrict__ idxs,
                         float* __restrict__ vals, const int* __restrict__ first,
                         int T) {
  int t = blockIdx.x * blockDim.x + threadIdx.x;
  if (t < NGRAM - 1 || t >= T) return;
  int s = slot[t];
  if (first[s] != t) return;  // not the leader for this slot

  u64 skey = 0ull;            // matches reference keys[] zero-init
  int stoks[TOP_M] = {0, 0, 0, 0};
  int scnts[TOP_M] = {0, 0, 0, 0};

  for (int j = t; j < T; ++j) {
    if (slot[j] != s) continue;
    u64 wk = key[j];
    bool key_match = (skey == wk);

    // ---- emission (reads pre-update state) ----
    int total = 0;
#pragma unroll
    for (int m = 0; m < TOP_M; ++m)
      if (stoks[m] > 0) total += scnts[m];
    bool emit = key_match && (total > 0);
    float denom = (float)total;
    if (denom < 1.0f) denom = 1.0f;
#pragma unroll
    for (int m = 0; m < TOP_M; ++m) {
      bool ok = emit && (stoks[m] > 0) && (scnts[m] > 0);
      float p = (float)scnts[m] / denom;
      if (p < 1e-9f) p = 1e-9f;
      vals[j * TOP_M + m] = ok ? (logf(p) * 0.5f) : 0.0f;
      idxs[j * TOP_M + m] = ok ? (stoks[m] - 1) : 0;
    }

    // ---- update (only when a next token exists) ----
    if (j < T - 1) {
      if (!key_match) {
#pragma unroll
        for (int m = 0; m < TOP_M; ++m) { stoks[m] = 0; scnts[m] = 0; }
      }
      skey = wk;
      int nt = ntp1[j];
      int ins = -1;
#pragma unroll
      for (int m = 0; m < TOP_M; ++m)
        if (ins < 0 && stoks[m] == nt) ins = m;  // first match
      int new_cnt;
      if (ins >= 0) {
        new_cnt = scnts[ins] + 1;
        if (new_cnt > 65535) new_cnt = 65535;
      } else {
#pragma unroll
        for (int m = 0; m < TOP_M; ++m)
          if (ins < 0 && stoks[m] == 0) ins = m;  // first empty
        if (ins < 0) {                            // first argmin(cnts)
          int mn = scnts[0];
          ins = 0;
#pragma unroll
          for (int m = 1; m < TOP_M; ++m)
            if (scnts[m] < mn) { mn = scnts[m]; ins = m; }
        }
        new_cnt = 1;
      }
      stoks[ins] = nt;
      scnts[ins] = new_cnt;
    }
  }
}

// ---------- K4: apply sparse deltas ----------
__global__ void k_apply(const int* __restrict__ idxs, const float* __restrict__ vals,
                        float* __restrict__ out, int T, long long V) {
  int i = blockIdx.x * blockDim.x + threadIdx.x;
  if (i >= T * TOP_M) return;
  float v = vals[i];
  if (v != 0.0f) {
    int t = i >> 2;
    int idx = idxs[i];
    atomicAdd(&out[(long long)t * V + idx], v);
  }
}

extern "C" void kernel_launch(void* const* d_in, const int* in_sizes, int n_in,
                              void* d_out, int out_size, void* d_ws, size_t ws_size,
                              hipStream_t stream) {
  const float* logits = (const float*)d_in[0];
  const int* ids = (const int*)d_in[1];
  float* out = (float*)d_out;

  int T = in_sizes[1];
  long long V = (long long)in_sizes[0] / T;

  // workspace layout
  char* ws = (char*)d_ws;
  size_t off = 0;
  int* first = (int*)(ws + off); off += (size_t)TABLE * 4;
  int* slot  = (int*)(ws + off); off += (size_t)T * 4;
  int* ntp1  = (int*)(ws + off); off += (size_t)T * 4;
  off = (off + 7) & ~(size_t)7;
  u64* key   = (u64*)(ws + off); off += (size_t)T * 8;
  int* idxs  = (int*)(ws + off); off += (size_t)T * TOP_M * 4;
  float* vals = (float*)(ws + off);

  k_init_first<<<TABLE / BT, BT, 0, stream>>>(first);
  k_prep<<<(T + BT - 1) / BT, BT, 0, stream>>>(ids, slot, key, ntp1, idxs, vals,
                                               first, T);
  k_chains<<<(T + BT - 1) / BT, BT, 0, stream>>>(slot, key, ntp1, idxs, vals,
                                                 first, T);
  k_copy<<<8192, BT, 0, stream>>>(logits, out, (long long)out_size);
  k_apply<<<(T * TOP_M + BT - 1) / BT, BT, 0, stream>>>(idxs, vals, out, T, V);
}